// TextureMapper_54924041782037
// MI455X (gfx1250) — compile-verified
//
#include <hip/hip_runtime.h>
#include <cstdint>

#define C_ 16
#define SH_START 3

__device__ __forceinline__ void tap_accum(float* acc, const float* __restrict__ tex,
                                          int T, int yi, int xi, float w) {
  // zero-padding: skip out-of-range taps (contribution is zero)
  if ((unsigned)yi < (unsigned)T && (unsigned)xi < (unsigned)T) {
    const float4* p = reinterpret_cast<const float4*>(tex + ((size_t)yi * (size_t)T + (size_t)xi) * C_);
#pragma unroll
    for (int q = 0; q < 4; ++q) {
      float4 t = p[q];   // global_load_b128, 64B-aligned
      acc[q * 4 + 0] = fmaf(w, t.x, acc[q * 4 + 0]);
      acc[q * 4 + 1] = fmaf(w, t.y, acc[q * 4 + 1]);
      acc[q * 4 + 2] = fmaf(w, t.z, acc[q * 4 + 2]);
      acc[q * 4 + 3] = fmaf(w, t.w, acc[q * 4 + 3]);
    }
  }
}

__device__ __forceinline__ void sample_level(float* acc, const float* __restrict__ tex,
                                             int T, float u, float v) {
  // matches ref: ix = u*T - 0.5 ; iy = (1-v)*T - 0.5 (y-flip folded in)
  float ix = fmaf(u, (float)T, -0.5f);
  float iy = fmaf(1.0f - v, (float)T, -0.5f);
  float fx0 = floorf(ix), fy0 = floorf(iy);
  int x0 = (int)fx0, y0 = (int)fy0;
  float wx1 = ix - fx0, wx0 = 1.0f - wx1;
  float wy1 = iy - fy0, wy0 = 1.0f - wy1;
  tap_accum(acc, tex, T, y0,     x0,     wy0 * wx0);
  tap_accum(acc, tex, T, y0,     x0 + 1, wy0 * wx1);
  tap_accum(acc, tex, T, y0 + 1, x0,     wy1 * wx0);
  tap_accum(acc, tex, T, y0 + 1, x0 + 1, wy1 * wx1);
}

__global__ __launch_bounds__(256)
void texmap_kernel(const float* __restrict__ uv,
                   const float* __restrict__ sh,
                   const float* __restrict__ t0,
                   const float* __restrict__ t1,
                   const float* __restrict__ t2,
                   const float* __restrict__ t3,
                   float* __restrict__ out) {
  __shared__ float s_uv[256 * 2];
  const int tid = threadIdx.x;
  const unsigned gid = blockIdx.x * 256u + (unsigned)tid;   // flat pixel index

  const float* shp = sh + (size_t)gid * 9u;
  __builtin_prefetch(shp, 0, 1);   // global_prefetch_b8

  float u, v;
#if defined(__gfx1250__)
  {
    // Async DMA this lane's 8B uv pair into LDS (CDNA5 async path, ASYNCcnt).
    // Low 32 bits of a generic shared pointer == LDS byte offset on AMDGPU.
    unsigned lds_off = (unsigned)(uintptr_t)(&s_uv[tid * 2]);
    unsigned voff    = gid * 8u;   // byte offset from uniform SGPR base
    asm volatile("global_load_async_to_lds_b64 %0, %1, %2"
                 :: "v"(lds_off), "v"(voff), "s"(uv)
                 : "memory");
    asm volatile("s_wait_asynccnt 0" ::: "memory");
  }
  u = s_uv[tid * 2 + 0];
  v = s_uv[tid * 2 + 1];
#else
  u = uv[(size_t)gid * 2 + 0];
  v = uv[(size_t)gid * 2 + 1];
#endif

  float acc[16];
#pragma unroll
  for (int i = 0; i < 16; ++i) acc[i] = 0.0f;

  sample_level(acc, t0, 1024, u, v);
  sample_level(acc, t1,  512, u, v);
  sample_level(acc, t2,  256, u, v);
  sample_level(acc, t3,  128, u, v);

  // mask where u == 0 (exact compare, as in reference)
  float msk = (u == 0.0f) ? 0.0f : 1.0f;
#pragma unroll
  for (int i = 0; i < 16; ++i) acc[i] *= msk;

  // SH multiply on channels [3, 12); sh is read-once -> NT loads
#pragma unroll
  for (int i = 0; i < 9; ++i)
    acc[SH_START + i] *= __builtin_nontemporal_load(shp + i);

  // NCHW scatter: gid = ((n*512 + y)*512 + x); per-channel stores are coalesced.
  const unsigned rem = gid & 0x3FFFFu;        // y*512 + x
  const unsigned n   = gid >> 18;             // batch
  float* ob = out + ((size_t)n << 22) + rem;  // n*C*H*W
#pragma unroll
  for (int c = 0; c < 16; ++c)
    __builtin_nontemporal_store(acc[c], ob + ((size_t)c << 18));  // th:NT streaming
}

extern "C" void kernel_launch(void* const* d_in, const int* in_sizes, int n_in,
                              void* d_out, int out_size, void* d_ws, size_t ws_size,
                              hipStream_t stream) {
  (void)in_sizes; (void)n_in; (void)out_size; (void)d_ws; (void)ws_size;
  const float* uv = (const float*)d_in[0];
  const float* sh = (const float*)d_in[1];
  const float* t0 = (const float*)d_in[2];
  const float* t1 = (const float*)d_in[3];
  const float* t2 = (const float*)d_in[4];
  const float* t3 = (const float*)d_in[5];
  float* out = (float*)d_out;

  const int total_pixels = 8 * 512 * 512;        // N*H*W = 2,097,152
  dim3 grid(total_pixels / 256), block(256);
  texmap_kernel<<<grid, block, 0, stream>>>(uv, sh, t0, t1, t2, t3, out);
}